// HopGCN_53893249630259
// MI455X (gfx1250) — compile-verified
//
#include <hip/hip_runtime.h>
#include <math.h>

// Problem constants (HopGCN reference): B=8, CIN=64, N=256, OUT=64, HEAD=4, DPH=16, DNUM=5
#define EPS_  1e-5f
#define NPIX  524288.0f   // B*N*N
#define XS    68          // LDS stride for X chunk [64 px][64 ch] (mult of 4 for float4, 68%64=4 for banks)
#define HS    34          // LDS stride for hpost  [px][32 ch]
#define ES    66          // LDS stride for e2 buf [px][64 ch]

typedef float v2f __attribute__((ext_vector_type(2)));
typedef float v8f __attribute__((ext_vector_type(8)));

__device__ __forceinline__ v8f zero8() {
  v8f z;
#pragma unroll
  for (int i = 0; i < 8; ++i) z[i] = 0.f;
  return z;
}

// CDNA5 fp32 WMMA: D(16x16) = A(16x4) * B(4x16) + C
__device__ __forceinline__ v8f wmma4(v2f a, v2f b, v8f c) {
  return __builtin_amdgcn_wmma_f32_16x16x4_f32(false, a, false, b, (short)0, c, false, false);
}

__device__ __forceinline__ float lrelu(float x) { return x < 0.f ? 0.2f * x : x; }

// ---------------------------------------------------------------------------
// K1: low-rank factors of first conv.
//   A [b][k][o] = sum_c W1[o,c]      * pf[b,c,k]
//   Bm[b][n][o] = sum_c (W1[o,64+c]-W1[o,c]) * pf[b,c,n]
// ---------------------------------------------------------------------------
__global__ __launch_bounds__(256) void k1_lowrank(
    const float* __restrict__ pf, const float* __restrict__ W1,
    float* __restrict__ At, float* __restrict__ Bt) {
  __shared__ float W1L[64 * 128];
  __shared__ float pfL[64 * 32];
  int b  = blockIdx.x;
  int kc = blockIdx.y * 32;
  int t  = threadIdx.x;
  for (int i = t; i < 64 * 128; i += 256) W1L[i] = W1[i];
  for (int i = t; i < 64 * 32; i += 256) {
    int c = i >> 5, kk = i & 31;
    pfL[i] = pf[((size_t)b * 64 + c) * 256 + kc + kk];
  }
  __syncthreads();
  int o = t & 63;
  int q = t >> 6;
  for (int kk = q * 8; kk < q * 8 + 8; ++kk) {
    float a = 0.f, bm = 0.f;
#pragma unroll
    for (int c = 0; c < 64; ++c) {
      float p  = pfL[c * 32 + kk];
      float wa = W1L[o * 128 + c];
      float wb = W1L[o * 128 + 64 + c];
      a  += wa * p;
      bm += (wb - wa) * p;
    }
    size_t idx = ((size_t)b * 256 + kc + kk) * 64 + o;
    At[idx] = a;
    Bt[idx] = bm;
  }
}

// ---------------------------------------------------------------------------
// K2: analytic BN1 stats.  y = A[b,o,k] + Bm[b,o,n] over (b,n,k).
// ---------------------------------------------------------------------------
__global__ __launch_bounds__(64) void k2_stats1(
    const float* __restrict__ At, const float* __restrict__ Bt,
    const float* __restrict__ g1, const float* __restrict__ b1,
    float* __restrict__ sc1) {
  int o = threadIdx.x;
  float SA = 0, SA2 = 0, SB = 0, SB2 = 0, cross = 0;
  for (int b = 0; b < 8; ++b) {
    float sa = 0, sa2 = 0, sb = 0, sb2 = 0;
    for (int k = 0; k < 256; ++k) {
      size_t idx = ((size_t)b * 256 + k) * 64 + o;
      float va = At[idx]; sa += va; sa2 += va * va;
      float vb = Bt[idx]; sb += vb; sb2 += vb * vb;
    }
    SA += sa; SA2 += sa2; SB += sb; SB2 += sb2; cross += sa * sb;
  }
  float m   = (SA + SB) * (1.0f / 2048.0f);
  float ey2 = (SA2 + SB2) * (1.0f / 2048.0f) + 2.0f * cross * (1.0f / 524288.0f);
  float v   = ey2 - m * m;
  float sc  = g1[o] * rsqrtf(v + EPS_);
  sc1[o]      = sc;
  sc1[64 + o] = b1[o] - m * sc;
}

// K2b: fold BN1 affine into the factors (in place): Asc = s*A ; Bsc = s*Bm + t
__global__ __launch_bounds__(256) void k2b_scale(
    float* __restrict__ At, float* __restrict__ Bt, const float* __restrict__ sc1) {
  int idx = blockIdx.x * 256 + threadIdx.x;  // 512 blocks -> 131072 elems
  int o = idx & 63;
  float sc = sc1[o], sh = sc1[64 + o];
  At[idx] = sc * At[idx];
  Bt[idx] = sc * Bt[idx] + sh;
}

// ---------------------------------------------------------------------------
// Shared per-row preamble: att1 logits + stable softmax denominators.
// ---------------------------------------------------------------------------
__device__ void row_pre(const float* __restrict__ Asc, const float* __restrict__ Bsc,
                        const float* __restrict__ a1, int b, int n,
                        float* Brow, float* sbuf, float* mx, float* isum) {
  int t = threadIdx.x;
  if (t < 64) Brow[t] = Bsc[((size_t)b * 256 + n) * 64 + t];
  __syncthreads();
#pragma unroll
  for (int rep = 0; rep < 2; ++rep) {
    int k = t + rep * 128;
    const float4* ar = reinterpret_cast<const float4*>(Asc + ((size_t)b * 256 + k) * 64);
    float s0 = 0.f, s1 = 0.f, s2 = 0.f, s3 = 0.f;
#pragma unroll
    for (int d = 0; d < 16; ++d) {
      float4 av = ar[d];
      float ad = a1[d];
      float e0 = lrelu(av.x + Brow[4 * d + 0]);
      float e1 = lrelu(av.y + Brow[4 * d + 1]);
      float e2 = lrelu(av.z + Brow[4 * d + 2]);
      float e3 = lrelu(av.w + Brow[4 * d + 3]);
      s0 += ad * lrelu(e0); s1 += ad * lrelu(e1);
      s2 += ad * lrelu(e2); s3 += ad * lrelu(e3);
    }
    sbuf[k * 4 + 0] = s0; sbuf[k * 4 + 1] = s1;
    sbuf[k * 4 + 2] = s2; sbuf[k * 4 + 3] = s3;
  }
  __syncthreads();
  int w = t >> 5, lane = t & 31;   // wave w handles head h = w
  float vals[8], m = -3.0e38f;
#pragma unroll
  for (int i = 0; i < 8; ++i) { vals[i] = sbuf[(lane + i * 32) * 4 + w]; m = fmaxf(m, vals[i]); }
#pragma unroll
  for (int d = 16; d >= 1; d >>= 1) m = fmaxf(m, __shfl_xor(m, d));
  float s = 0.f;
#pragma unroll
  for (int i = 0; i < 8; ++i) s += expf(vals[i] - m);
#pragma unroll
  for (int d = 16; d >= 1; d >>= 1) s += __shfl_xor(s, d);
  if (lane == 0) { mx[w] = m; isum[w] = 1.0f / s; }
  __syncthreads();
}

// Build 64-px chunk of X = e1_new = att1 * lrelu(z) into LDS [px][ch], stride XS.
__device__ void build_chunk(const float* __restrict__ Asc, int b, int c0,
                            const float* Brow, const float* sbuf,
                            const float* mx, const float* isum, float* Xl) {
  int t  = threadIdx.x;
  int p  = t & 63;
  int hh = t >> 6;
  int k  = c0 + p;
  float att[4];
#pragma unroll
  for (int h = 0; h < 4; ++h) att[h] = expf(sbuf[k * 4 + h] - mx[h]) * isum[h];
  const float4* ar = reinterpret_cast<const float4*>(Asc + ((size_t)b * 256 + k) * 64);
  float* xr = Xl + p * XS;
#pragma unroll
  for (int d = hh * 8; d < hh * 8 + 8; ++d) {
    float4 av = ar[d];
    float4 xv;
    xv.x = att[0] * lrelu(av.x + Brow[4 * d + 0]);
    xv.y = att[1] * lrelu(av.y + Brow[4 * d + 1]);
    xv.z = att[2] * lrelu(av.z + Brow[4 * d + 2]);
    xv.w = att[3] * lrelu(av.w + Brow[4 * d + 3]);
    *reinterpret_cast<float4*>(xr + 4 * d) = xv;
  }
}

// ---------------------------------------------------------------------------
// K3: global Gram G = sum_px x x^T (64x64) and s = sum_px x, via fp32 WMMA.
// Block = (b, 8 rows), 4 waves = 4 M-strips.
// ---------------------------------------------------------------------------
__global__ __launch_bounds__(128) void k3_gram(
    const float* __restrict__ Asc, const float* __restrict__ Bsc,
    const float* __restrict__ a1,
    float* __restrict__ Gpart, float* __restrict__ spart) {
  __shared__ alignas(16) float Xl[64 * XS];
  __shared__ float Brow[64], sbuf[256 * 4], mx[4], isum[4];
  __shared__ float sp2[2][64], sacc[64];
  int t = threadIdx.x, blk = blockIdx.x;
  int b = blk >> 5, ng = blk & 31;
  int w = t >> 5, lane = t & 31, half = lane >> 4, lm = lane & 15;
  v8f acc[4];
#pragma unroll
  for (int nt = 0; nt < 4; ++nt) acc[nt] = zero8();
  if (t < 64) sacc[t] = 0.f;
  __syncthreads();
  for (int r8 = 0; r8 < 8; ++r8) {
    int n = ng * 8 + r8;
    row_pre(Asc, Bsc, a1, b, n, Brow, sbuf, mx, isum);
    for (int cc = 0; cc < 4; ++cc) {
      build_chunk(Asc, b, cc * 64, Brow, sbuf, mx, isum, Xl);
      __syncthreads();
      {  // channel-sum partials
        int oo = t & 63, q = t >> 6;
        float sv = 0.f;
        for (int p = q * 32; p < q * 32 + 32; ++p) sv += Xl[p * XS + oo];
        sp2[q][oo] = sv;
      }
#pragma unroll
      for (int j = 0; j < 16; ++j) {  // K chunks over 64 px
        int kk = 4 * j + 2 * half;
        v2f a; a.x = Xl[kk * XS + 16 * w + lm]; a.y = Xl[(kk + 1) * XS + 16 * w + lm];
#pragma unroll
        for (int nt = 0; nt < 4; ++nt) {
          v2f bb; bb.x = Xl[kk * XS + 16 * nt + lm]; bb.y = Xl[(kk + 1) * XS + 16 * nt + lm];
          acc[nt] = wmma4(a, bb, acc[nt]);
        }
      }
      __syncthreads();
      if (t < 64) sacc[t] += sp2[0][t] + sp2[1][t];
      __syncthreads();
    }
  }
#pragma unroll
  for (int nt = 0; nt < 4; ++nt)
#pragma unroll
    for (int r = 0; r < 8; ++r)
      Gpart[(size_t)blk * 4096 + (16 * w + 8 * half + r) * 64 + 16 * nt + lm] = acc[nt][r];
  if (t < 64) spart[blk * 64 + t] = sacc[t];
}

// K4: reduce Gram/sum, derive BN params of Wh1-block (32ch) and W2-block (64ch)
__global__ __launch_bounds__(256) void k4_bn24(
    const float* __restrict__ Gpart, const float* __restrict__ spart,
    const float* __restrict__ Wh1, const float* __restrict__ gh1, const float* __restrict__ bh1,
    const float* __restrict__ W2, const float* __restrict__ g2, const float* __restrict__ b2,
    float* __restrict__ sc24) {
  __shared__ float GL[4096];
  __shared__ float sL[64];
  int t = threadIdx.x;
  for (int e = t; e < 4096; e += 256) {
    float acc = 0.f;
    for (int p = 0; p < 256; ++p) acc += Gpart[(size_t)p * 4096 + e];
    GL[e] = acc;
  }
  if (t < 64) {
    float acc = 0.f;
    for (int p = 0; p < 256; ++p) acc += spart[p * 64 + t];
    sL[t] = acc;
  }
  __syncthreads();
  const float inv = 1.0f / NPIX;
  if (t < 32) {
    const float* w = Wh1 + t * 64;
    float mean = 0.f;
    for (int i = 0; i < 64; ++i) mean += w[i] * sL[i];
    mean *= inv;
    float q = 0.f;
    for (int i = 0; i < 64; ++i) {
      float r = 0.f;
      for (int j = 0; j < 64; ++j) r += GL[i * 64 + j] * w[j];
      q += w[i] * r;
    }
    float var = q * inv - mean * mean;
    float sc = gh1[t] * rsqrtf(var + EPS_);
    sc24[t] = sc; sc24[32 + t] = bh1[t] - mean * sc;
  } else if (t < 96) {
    int o = t - 32;
    const float* w = W2 + o * 64;
    float mean = 0.f;
    for (int i = 0; i < 64; ++i) mean += w[i] * sL[i];
    mean *= inv;
    float q = 0.f;
    for (int i = 0; i < 64; ++i) {
      float r = 0.f;
      for (int j = 0; j < 64; ++j) r += GL[i * 64 + j] * w[j];
      q += w[i] * r;
    }
    float var = q * inv - mean * mean;
    float sc = g2[o] * rsqrtf(var + EPS_);
    sc24[64 + o] = sc; sc24[128 + o] = b2[o] - mean * sc;
  }
}

// ---------------------------------------------------------------------------
// K5: Gram/sum of hpost = lrelu(bn2(Wh1 @ X))  (32x32) for the hop-logits BN.
// ---------------------------------------------------------------------------
__global__ __launch_bounds__(128) void k5_hopstats(
    const float* __restrict__ Asc, const float* __restrict__ Bsc,
    const float* __restrict__ a1, const float* __restrict__ Wh1,
    const float* __restrict__ sc24,
    float* __restrict__ Ghpart, float* __restrict__ shpart) {
  __shared__ alignas(16) float Xl[64 * XS];
  __shared__ float Brow[64], sbuf[256 * 4], mx[4], isum[4];
  __shared__ float hp[64 * HS];
  __shared__ float shp4[4][32], shacc[32];
  int t = threadIdx.x, blk = blockIdx.x;
  int b = blk >> 5, ng = blk & 31;
  int w = t >> 5, lane = t & 31, half = lane >> 4, lm = lane & 15;
  int mt2 = w >> 1, nt2 = w & 1;
  v8f acch = zero8();
  if (t < 32) shacc[t] = 0.f;
  __syncthreads();
  for (int r8 = 0; r8 < 8; ++r8) {
    int n = ng * 8 + r8;
    row_pre(Asc, Bsc, a1, b, n, Brow, sbuf, mx, isum);
    for (int cc = 0; cc < 4; ++cc) {
      build_chunk(Asc, b, cc * 64, Brow, sbuf, mx, isum, Xl);
      __syncthreads();
      int lo = w * 16;   // wave's 16-px tile within chunk
#pragma unroll
      for (int mt = 0; mt < 2; ++mt) {
        v8f acc = zero8();
#pragma unroll
        for (int j = 0; j < 16; ++j) {
          int kk = 4 * j + 2 * half;
          v2f a  = *reinterpret_cast<const v2f*>(Wh1 + (16 * mt + lm) * 64 + kk);
          v2f bb = *reinterpret_cast<const v2f*>(Xl + (lo + lm) * XS + kk);
          acc = wmma4(a, bb, acc);
        }
#pragma unroll
        for (int r = 0; r < 8; ++r) {
          int ch = 16 * mt + 8 * half + r;
          hp[(lo + lm) * HS + ch] = lrelu(sc24[ch] * acc[r] + sc24[32 + ch]);
        }
      }
      __syncthreads();
      {  // s_h partials
        int o2 = t & 31, q = t >> 5;
        float sv = 0.f;
        for (int p = q * 16; p < q * 16 + 16; ++p) sv += hp[p * HS + o2];
        shp4[q][o2] = sv;
      }
#pragma unroll
      for (int j = 0; j < 16; ++j) {  // Gram over 64 px
        int kk = 4 * j + 2 * half;
        v2f a;  a.x  = hp[kk * HS + 16 * mt2 + lm]; a.y  = hp[(kk + 1) * HS + 16 * mt2 + lm];
        v2f bb; bb.x = hp[kk * HS + 16 * nt2 + lm]; bb.y = hp[(kk + 1) * HS + 16 * nt2 + lm];
        acch = wmma4(a, bb, acch);
      }
      __syncthreads();
      if (t < 32) shacc[t] += shp4[0][t] + shp4[1][t] + shp4[2][t] + shp4[3][t];
      __syncthreads();
    }
  }
#pragma unroll
  for (int r = 0; r < 8; ++r)
    Ghpart[(size_t)blk * 1024 + (16 * mt2 + 8 * half + r) * 32 + 16 * nt2 + lm] = acch[r];
  if (t < 32) shpart[blk * 32 + t] = shacc[t];
}

// K6: BN params for hop logits (5 channels)
__global__ __launch_bounds__(256) void k6_bn3(
    const float* __restrict__ Ghpart, const float* __restrict__ shpart,
    const float* __restrict__ Wh2, const float* __restrict__ gh2, const float* __restrict__ bh2,
    float* __restrict__ sc3) {
  __shared__ float GL[1024], sL[32];
  int t = threadIdx.x;
  for (int e = t; e < 1024; e += 256) {
    float a = 0.f;
    for (int p = 0; p < 256; ++p) a += Ghpart[(size_t)p * 1024 + e];
    GL[e] = a;
  }
  if (t < 32) {
    float a = 0.f;
    for (int p = 0; p < 256; ++p) a += shpart[p * 32 + t];
    sL[t] = a;
  }
  __syncthreads();
  if (t < 5) {
    const float* w = Wh2 + t * 32;
    float mean = 0.f;
    for (int i = 0; i < 32; ++i) mean += w[i] * sL[i];
    mean *= (1.0f / NPIX);
    float q = 0.f;
    for (int i = 0; i < 32; ++i) {
      float r = 0.f;
      for (int j = 0; j < 32; ++j) r += GL[i * 32 + j] * w[j];
      q += w[i] * r;
    }
    float var = q * (1.0f / NPIX) - mean * mean;
    float sc = gh2[t] * rsqrtf(var + EPS_);
    sc3[t] = sc; sc3[8 + t] = bh2[t] - mean * sc;
  }
}

// ---------------------------------------------------------------------------
// K7: final fused pass -> hop_logits output + argmax/gauss + att2 + g_out
// ---------------------------------------------------------------------------
__global__ __launch_bounds__(128) void k7_final(
    const float* __restrict__ Asc, const float* __restrict__ Bsc,
    const float* __restrict__ a1, const float* __restrict__ a2,
    const float* __restrict__ Wh1, const float* __restrict__ Wh2,
    const float* __restrict__ W2,
    const float* __restrict__ sc24, const float* __restrict__ sc3,
    float* __restrict__ out) {
  __shared__ alignas(16) float Xl[64 * XS];
  __shared__ float Brow[64], sbuf[256 * 4], mx[4], isum[4];
  __shared__ float hpW[4 * 16 * HS];
  __shared__ float e2b[4 * 16 * ES];
  __shared__ float s2buf[256 * 4];
  __shared__ float gaussL[256];
  __shared__ float mx2[4], isum2[4];
  __shared__ float gs2[64 * 4];
  int t = threadIdx.x, blk = blockIdx.x;
  int b = blk >> 5, ng = blk & 31;
  int w = t >> 5, lane = t & 31, half = lane >> 4, lm = lane & 15;
  float* hp = hpW + w * 16 * HS;
  float* eb = e2b + w * 16 * ES;
  float* g_out  = out;
  float* logits = out + 131072;

  for (int r8 = 0; r8 < 8; ++r8) {
    int n = ng * 8 + r8;
    row_pre(Asc, Bsc, a1, b, n, Brow, sbuf, mx, isum);
    // ---- phase A: logits, gauss, att2 logits ----
    for (int cc = 0; cc < 4; ++cc) {
      build_chunk(Asc, b, cc * 64, Brow, sbuf, mx, isum, Xl);
      __syncthreads();
      int lo = w * 16;
      int kb = cc * 64 + lo;
#pragma unroll
      for (int mt = 0; mt < 2; ++mt) {  // y_h = Wh1 @ X -> hpost
        v8f acc = zero8();
#pragma unroll
        for (int j = 0; j < 16; ++j) {
          int kk = 4 * j + 2 * half;
          v2f a  = *reinterpret_cast<const v2f*>(Wh1 + (16 * mt + lm) * 64 + kk);
          v2f bb = *reinterpret_cast<const v2f*>(Xl + (lo + lm) * XS + kk);
          acc = wmma4(a, bb, acc);
        }
#pragma unroll
        for (int r = 0; r < 8; ++r) {
          int ch = 16 * mt + 8 * half + r;
          hp[lm * HS + ch] = lrelu(sc24[ch] * acc[r] + sc24[32 + ch]);
        }
      }
      __syncthreads();
      if (lane < 16) {  // hop logits + argmax + gauss
        int k = kb + lane;
        float best = -3.0e38f; int bi = 0;
#pragma unroll
        for (int c = 0; c < 5; ++c) {
          float s = 0.f;
#pragma unroll
          for (int o2 = 0; o2 < 32; ++o2) s += Wh2[c * 32 + o2] * hp[lane * HS + o2];
          float lg = lrelu(sc3[c] * s + sc3[8 + c]);
          logits[(((size_t)b * 5 + c) * 256 + n) * 256 + k] = lg;
          if (lg > best) { best = lg; bi = c; }
        }
        gaussL[k] = expf(-0.5f * (float)(bi * bi)) * 0.3989422804014327f;
      }
#pragma unroll
      for (int mt = 0; mt < 4; ++mt) {  // y2 = W2 @ X -> e2
        v8f acc = zero8();
#pragma unroll
        for (int j = 0; j < 16; ++j) {
          int kk = 4 * j + 2 * half;
          v2f a  = *reinterpret_cast<const v2f*>(W2 + (16 * mt + lm) * 64 + kk);
          v2f bb = *reinterpret_cast<const v2f*>(Xl + (lo + lm) * XS + kk);
          acc = wmma4(a, bb, acc);
        }
#pragma unroll
        for (int r = 0; r < 8; ++r) {
          int ch = 16 * mt + 8 * half + r;
          eb[lm * ES + ch] = lrelu(sc24[64 + ch] * acc[r] + sc24[128 + ch]);
        }
      }
      __syncthreads();
      {  // att2 logits: s2[h,k] = sum_d a2[d] * lrelu(gauss * e2[4d+h])
        int k = kb + lm;
        float gv = gaussL[k];
#pragma unroll
        for (int hh = 0; hh < 2; ++hh) {
          int h = 2 * half + hh;
          float s = 0.f;
#pragma unroll
          for (int d = 0; d < 16; ++d) s += a2[d] * lrelu(gv * eb[lm * ES + 4 * d + h]);
          s2buf[k * 4 + h] = s;
        }
      }
      __syncthreads();
    }
    // ---- att2 softmax denominators (wave w handles head w) ----
    {
      float vals[8], m = -3.0e38f;
#pragma unroll
      for (int i = 0; i < 8; ++i) { vals[i] = s2buf[(lane + i * 32) * 4 + w]; m = fmaxf(m, vals[i]); }
#pragma unroll
      for (int d = 16; d >= 1; d >>= 1) m = fmaxf(m, __shfl_xor(m, d));
      float s = 0.f;
#pragma unroll
      for (int i = 0; i < 8; ++i) s += expf(vals[i] - m);
#pragma unroll
      for (int d = 16; d >= 1; d >>= 1) s += __shfl_xor(s, d);
      if (lane == 0) { mx2[w] = m; isum2[w] = 1.0f / s; }
    }
    __syncthreads();
    // ---- phase B: g_out = mean_k(att2 * e2) ----
    float gsum[4][8];
#pragma unroll
    for (int mt = 0; mt < 4; ++mt)
#pragma unroll
      for (int r = 0; r < 8; ++r) gsum[mt][r] = 0.f;
    for (int cc = 0; cc < 4; ++cc) {
      __syncthreads();
      build_chunk(Asc, b, cc * 64, Brow, sbuf, mx, isum, Xl);
      __syncthreads();
      int lo = w * 16;
      int k = cc * 64 + lo + lm;
      float att2v[4];
#pragma unroll
      for (int h = 0; h < 4; ++h) att2v[h] = expf(s2buf[k * 4 + h] - mx2[h]) * isum2[h];
#pragma unroll
      for (int mt = 0; mt < 4; ++mt) {
        v8f acc = zero8();
#pragma unroll
        for (int j = 0; j < 16; ++j) {
          int kk = 4 * j + 2 * half;
          v2f a  = *reinterpret_cast<const v2f*>(W2 + (16 * mt + lm) * 64 + kk);
          v2f bb = *reinterpret_cast<const v2f*>(Xl + (lo + lm) * XS + kk);
          acc = wmma4(a, bb, acc);
        }
#pragma unroll
        for (int r = 0; r < 8; ++r) {
          int ch = 16 * mt + 8 * half + r;
          float e2v = lrelu(sc24[64 + ch] * acc[r] + sc24[128 + ch]);
          gsum[mt][r] += att2v[r & 3] * e2v;  // head = ch % 4 = r % 4
        }
      }
    }
#pragma unroll
    for (int mt = 0; mt < 4; ++mt)
#pragma unroll
      for (int r = 0; r < 8; ++r) {
        float v = gsum[mt][r];
#pragma unroll
        for (int d = 1; d < 16; d <<= 1) v += __shfl_xor(v, d);
        if (lm == 0) gs2[(16 * mt + 8 * half + r) * 4 + w] = v;
      }
    __syncthreads();
    if (t < 64) {
      float g = gs2[t * 4 + 0] + gs2[t * 4 + 1] + gs2[t * 4 + 2] + gs2[t * 4 + 3];
      g_out[((size_t)b * 64 + t) * 256 + n] = g * (1.0f / 256.0f);
    }
    __syncthreads();
  }
}

// ---------------------------------------------------------------------------
extern "C" void kernel_launch(void* const* d_in, const int* in_sizes, int n_in,
                              void* d_out, int out_size, void* d_ws, size_t ws_size,
                              hipStream_t stream) {
  (void)in_sizes; (void)n_in; (void)out_size; (void)ws_size;
  const float* pf  = (const float*)d_in[0];
  const float* W1  = (const float*)d_in[1];
  const float* g1  = (const float*)d_in[2];
  const float* b1  = (const float*)d_in[3];
  const float* Wh1 = (const float*)d_in[4];
  const float* gh1 = (const float*)d_in[5];
  const float* bh1 = (const float*)d_in[6];
  const float* Wh2 = (const float*)d_in[7];
  const float* gh2 = (const float*)d_in[8];
  const float* bh2 = (const float*)d_in[9];
  const float* W2  = (const float*)d_in[10];
  const float* g2  = (const float*)d_in[11];
  const float* b2  = (const float*)d_in[12];
  const float* a1  = (const float*)d_in[13];
  const float* a2  = (const float*)d_in[14];
  float* ws = (float*)d_ws;
  float* At    = ws;                 // 131072, becomes Asc after K2b
  float* Bt    = ws + 131072;        // 131072, becomes Bsc after K2b
  float* sc1   = ws + 262144;        // 128
  float* Gpart = ws + 262272;        // 256*4096
  float* spart = ws + 1310848;       // 256*64
  float* sc24  = ws + 1327232;       // 192
  float* Ghp   = ws + 1327424;       // 256*1024
  float* shp   = ws + 1589568;       // 256*32
  float* sc3   = ws + 1597760;       // 16
  float* outp  = (float*)d_out;

  k1_lowrank<<<dim3(8, 8), dim3(256), 0, stream>>>(pf, W1, At, Bt);
  k2_stats1<<<dim3(1), dim3(64), 0, stream>>>(At, Bt, g1, b1, sc1);
  k2b_scale<<<dim3(512), dim3(256), 0, stream>>>(At, Bt, sc1);
  k3_gram<<<dim3(256), dim3(128), 0, stream>>>(At, Bt, a1, Gpart, spart);
  k4_bn24<<<dim3(1), dim3(256), 0, stream>>>(Gpart, spart, Wh1, gh1, bh1, W2, g2, b2, sc24);
  k5_hopstats<<<dim3(256), dim3(128), 0, stream>>>(At, Bt, a1, Wh1, sc24, Ghp, shp);
  k6_bn3<<<dim3(1), dim3(256), 0, stream>>>(Ghp, shp, Wh2, gh2, bh2, sc3);
  k7_final<<<dim3(256), dim3(128), 0, stream>>>(At, Bt, a1, a2, Wh1, Wh2, W2, sc24, sc3, outp);
}